// VectorQuantize_68289980006516
// MI455X (gfx1250) — compile-verified
//
#include <hip/hip_runtime.h>
#include <hip/hip_bf16.h>
#include <math.h>

typedef __attribute__((ext_vector_type(2))) float v2f;
typedef __attribute__((ext_vector_type(8))) float v8f;

#define N_ROWS     32768
#define DIM        64
#define KCODES     1024
#define CB_CHUNK   256          // codes staged in LDS at a time
#define LDS_STRIDE 68           // padded row (floats): bank shift of 4/row -> conflict-free
#define TOTAL_ELEM (N_ROWS * DIM)

// ---------------------------------------------------------------------------
// Kernel 0: zero workspace (hist[1024] + sqsum[1])
// ---------------------------------------------------------------------------
__global__ void vq_zero_kernel(float* __restrict__ ws) {
    int i = blockIdx.x * blockDim.x + threadIdx.x;
    if (i < KCODES + 1) ws[i] = 0.0f;
}

// ---------------------------------------------------------------------------
// Kernel 1: distances via V_WMMA_F32_16X16X4_F32, argmin, gather, ST output,
//           squared-error accumulation, histogram.
// Block = 256 threads (8 waves), each wave owns 16 rows. Grid = 256 blocks.
// ---------------------------------------------------------------------------
__global__ __launch_bounds__(256) void vq_main_kernel(
    const float* __restrict__ X,      // (32768, 64)
    const float* __restrict__ E,      // (1024, 64)
    float* __restrict__ out,          // (32768, 64) quantize_st
    float* __restrict__ hist,         // (1024,)
    float* __restrict__ sqsum)        // (1,)
{
    __shared__ float lds_cb[CB_CHUNK * LDS_STRIDE];   // 69,632 B
    __shared__ float lds_en[CB_CHUNK];                // ||e||^2 per staged code

    const int tid  = threadIdx.x;
    const int lane = tid & 31;
    const int wave = tid >> 5;
    const int half = lane >> 4;          // 0: lanes 0-15, 1: lanes 16-31
    const int j    = lane & 15;
    const int row0 = (blockIdx.x * 8 + wave) * 16;

    // ---- Load A (16 rows x 64 dims) into WMMA 16x16x4 layout: 16 chunks ----
    // lane L holds x[row0 + (L&15)][4*kc + 2*half + {0,1}]
    v2f A[16];
    const float* xrow = X + (size_t)(row0 + j) * DIM;
    float xn_part = 0.0f;
    #pragma unroll
    for (int kc = 0; kc < 16; ++kc) {
        float2 t = *(const float2*)(xrow + kc * 4 + half * 2);
        A[kc].x = t.x; A[kc].y = t.y;
        xn_part += t.x * t.x + t.y * t.y;
    }
    // full ||x||^2 of row (row0+j): lane pair (j, j+16) holds the two halves
    float xnorm_row = xn_part + __shfl_xor(xn_part, 16, 32);
    // xn[v] = ||x||^2 of row (8*half + v) -- matches C-matrix M mapping
    float xn[8];
    #pragma unroll
    for (int v = 0; v < 8; ++v)
        xn[v] = __shfl(xnorm_row, half * 8 + v, 32);

    float best_d[8];
    int   best_i[8];
    #pragma unroll
    for (int v = 0; v < 8; ++v) { best_d[v] = 3.4e38f; best_i[v] = 0; }

    // ---- Sweep codebook in 4 chunks of 256 codes ----
    for (int chunk = 0; chunk < 4; ++chunk) {
        __syncthreads();   // everyone done with previous chunk's LDS
        // Coalesced stage: 4096 float4s; thread strides by 256
        #pragma unroll
        for (int it = 0; it < 16; ++it) {
            int f  = it * 256 + tid;      // float4 index within chunk
            int r  = f >> 4;              // code row within chunk
            int d4 = f & 15;              // float4 within row
            float4 t = *(const float4*)(E + (size_t)(chunk * CB_CHUNK + r) * DIM + d4 * 4);
            *(float4*)(lds_cb + r * LDS_STRIDE + d4 * 4) = t;
        }
        __syncthreads();
        // ||e||^2 of staged code `tid`
        {
            float en = 0.0f;
            const float* rr = lds_cb + tid * LDS_STRIDE;
            #pragma unroll
            for (int d = 0; d < DIM; ++d) en += rr[d] * rr[d];
            lds_en[tid] = en;
        }
        __syncthreads();

        // 16 column-tiles of 16 codes each
        for (int t16 = 0; t16 < 16; ++t16) {
            const int cloc = t16 * 16;
            v8f acc = {0.f,0.f,0.f,0.f,0.f,0.f,0.f,0.f};
            const float* brow = lds_cb + (cloc + j) * LDS_STRIDE + half * 2;
            #pragma unroll
            for (int kc = 0; kc < 16; ++kc) {
                float2 b2 = *(const float2*)(brow + kc * 4);
                v2f B; B.x = b2.x; B.y = b2.y;
                acc = __builtin_amdgcn_wmma_f32_16x16x4_f32(
                        /*neg_a=*/false, A[kc], /*neg_b=*/false, B,
                        /*c_mod=*/(short)0, acc,
                        /*reuse_a=*/false, /*reuse_b=*/false);
            }
            const float en_n  = lds_en[cloc + j];
            const int   cbase = chunk * CB_CHUNK + cloc + j;   // this lane's column code
            #pragma unroll
            for (int v = 0; v < 8; ++v) {
                float dist = xn[v] - 2.0f * acc[v] + en_n;
                if (dist < best_d[v]) { best_d[v] = dist; best_i[v] = cbase; }
            }
        }
    }

    // ---- Argmin across the 16 lanes of each half (first-min tie-break) ----
    #pragma unroll
    for (int off = 1; off < 16; off <<= 1) {
        #pragma unroll
        for (int v = 0; v < 8; ++v) {
            float od = __shfl_xor(best_d[v], off, 32);
            int   oi = __shfl_xor(best_i[v], off, 32);
            if (od < best_d[v] || (od == best_d[v] && oi < best_i[v])) {
                best_d[v] = od; best_i[v] = oi;
            }
        }
    }
    // Now every lane of a half holds, per v, the argmin of row (row0 + 8*half + v).

    // ---- Gather codes, write straight-through output, accumulate (q-x)^2 ----
    float sq = 0.0f;
    #pragma unroll
    for (int v = 0; v < 8; ++v) {
        const int row = row0 + half * 8 + v;
        const int idx = best_i[v];
        const float4 xv = *(const float4*)(X + (size_t)row * DIM + j * 4);
        const float4 qv = *(const float4*)(E + (size_t)idx * DIM + j * 4);
        float4 o;
        float dx = qv.x - xv.x, dy = qv.y - xv.y, dz = qv.z - xv.z, dw = qv.w - xv.w;
        o.x = xv.x + dx; o.y = xv.y + dy; o.z = xv.z + dz; o.w = xv.w + dw;
        *(float4*)(out + (size_t)row * DIM + j * 4) = o;
        sq += dx * dx + dy * dy + dz * dz + dw * dw;
    }

    // histogram: lanes j<8 in each half cover the 16 rows of this wave
    if (j < 8) atomicAdd(hist + best_i[j], 1.0f);

    // reduce squared error across wave, one atomic per wave
    #pragma unroll
    for (int off = 1; off < 32; off <<= 1) sq += __shfl_xor(sq, off, 32);
    if (lane == 0) atomicAdd(sqsum, sq);
}

// ---------------------------------------------------------------------------
// Kernel 2: finalize loss + perplexity into d_out tail
// ---------------------------------------------------------------------------
__global__ __launch_bounds__(256) void vq_finalize_kernel(
    const float* __restrict__ hist, const float* __restrict__ sqsum,
    float* __restrict__ out)
{
    __shared__ float red[256];
    const int t = threadIdx.x;
    float acc = 0.0f;
    for (int i = t; i < KCODES; i += 256) {
        float p = hist[i] * (1.0f / (float)N_ROWS);
        acc += p * logf(p + 1e-10f);
    }
    red[t] = acc;
    __syncthreads();
    for (int s = 128; s > 0; s >>= 1) {
        if (t < s) red[t] += red[t + s];
        __syncthreads();
    }
    if (t == 0) {
        // codebook_loss == commitment_loss numerically -> loss = 1.25 * MSE
        float mse = sqsum[0] / (float)TOTAL_ELEM;
        out[TOTAL_ELEM]     = 1.25f * mse;       // loss
        out[TOTAL_ELEM + 1] = expf(-red[0]);     // perplexity
    }
}

// ---------------------------------------------------------------------------
extern "C" void kernel_launch(void* const* d_in, const int* in_sizes, int n_in,
                              void* d_out, int out_size, void* d_ws, size_t ws_size,
                              hipStream_t stream) {
    const float* X = (const float*)d_in[0];   // inputs  (32,32,32,64) f32
    const float* E = (const float*)d_in[1];   // codebook (1024,64)    f32
    float* out  = (float*)d_out;
    float* ws   = (float*)d_ws;
    float* hist = ws;                         // 1024 floats
    float* sqs  = ws + KCODES;                // 1 float

    vq_zero_kernel<<<(KCODES + 1 + 255) / 256, 256, 0, stream>>>(ws);
    vq_main_kernel<<<N_ROWS / (16 * 8), 256, 0, stream>>>(X, E, out, hist, sqs);
    vq_finalize_kernel<<<1, 256, 0, stream>>>(hist, sqs, out);
}